// Actor_49890340110402
// MI455X (gfx1250) — compile-verified
//
#include <hip/hip_runtime.h>
#include <hip/hip_bf16.h>

// ---------------------------------------------------------------------------
// Problem constants (from reference)
// ---------------------------------------------------------------------------
#define DD       1024          // device_num
#define NB       16            // neighbors
#define HID      256           // hidden dim
#define IN_DIM   1067008       // D*D + D + D*N + D
#define OUT_DIM  17408         // D*(N+1)
#define REP_END  1048576       // D*D
#define TASK_END 1049600       // + D
#define TOPO_END 1065984       // + D*N
#define CHUNK    8192          // K-chunk for layer-1 streaming
#define NCHUNK   131           // ceil(IN_DIM / CHUNK); last chunk = 2048

typedef float v2f __attribute__((ext_vector_type(2)));
typedef float v4f __attribute__((ext_vector_type(4)));
typedef float v8f __attribute__((ext_vector_type(8)));

// ---------------------------------------------------------------------------
// Kernel 1: layer-1 GEMV partials.  HBM-bound: streams W1 (1.09 GB) once with
// non-temporal b128 loads (read-once, don't pollute the 192MB L2).  x is
// gathered on the fly into LDS.  Deterministic per-(chunk,row) partials.
// grid = (NCHUNK, 8), block = 256 (8 waves; 4 rows/wave; 32 rows/block)
// ---------------------------------------------------------------------------
__global__ void k_layer1(const float* __restrict__ rep,
                         const float* __restrict__ task,
                         const int*   __restrict__ topo,
                         const float* __restrict__ rmask,
                         const float* __restrict__ W1,
                         float*       __restrict__ partial)
{
    __shared__ v4f sx4[CHUNK / 4];
    float* sx = reinterpret_cast<float*>(sx4);

    const int c0 = blockIdx.x * CHUNK;
    int kend = IN_DIM - c0;
    if (kend > CHUNK) kend = CHUNK;

    // Cooperative gather of x[c0 .. c0+CHUNK) into LDS.
    for (int t = threadIdx.x; t < CHUNK; t += 256) {
        const int k = c0 + t;
        float v = 0.0f;
        if (k < REP_END)       v = rep[k];
        else if (k < TASK_END) v = task[k - REP_END];
        else if (k < TOPO_END) v = (float)topo[k - TASK_END];
        else if (k < IN_DIM)   v = rmask[k - TOPO_END];
        sx[t] = v;
    }
    __syncthreads();

    const int lane = threadIdx.x & 31;
    const int wave = threadIdx.x >> 5;
    const int row0 = blockIdx.y * 32 + wave * 4;
    const int nIter = kend >> 7;               // 128 floats per wave-iteration

    const float* wbase = W1 + (size_t)row0 * IN_DIM + c0;

    float acc0 = 0.f, acc1 = 0.f, acc2 = 0.f, acc3 = 0.f;
    for (int it = 0; it < nIter; ++it) {
        const int idx = it * 32 + lane;
        const v4f xv = sx4[idx];
        const v4f* p0 = reinterpret_cast<const v4f*>(wbase) + idx;
        const v4f* p1 = reinterpret_cast<const v4f*>(wbase + (size_t)1 * IN_DIM) + idx;
        const v4f* p2 = reinterpret_cast<const v4f*>(wbase + (size_t)2 * IN_DIM) + idx;
        const v4f* p3 = reinterpret_cast<const v4f*>(wbase + (size_t)3 * IN_DIM) + idx;
        const v4f w0 = __builtin_nontemporal_load(p0);
        const v4f w1 = __builtin_nontemporal_load(p1);
        const v4f w2 = __builtin_nontemporal_load(p2);
        const v4f w3 = __builtin_nontemporal_load(p3);
        acc0 += w0[0]*xv[0] + w0[1]*xv[1] + w0[2]*xv[2] + w0[3]*xv[3];
        acc1 += w1[0]*xv[0] + w1[1]*xv[1] + w1[2]*xv[2] + w1[3]*xv[3];
        acc2 += w2[0]*xv[0] + w2[1]*xv[1] + w2[2]*xv[2] + w2[3]*xv[3];
        acc3 += w3[0]*xv[0] + w3[1]*xv[1] + w3[2]*xv[2] + w3[3]*xv[3];
    }

    // wave32 reduction
    #pragma unroll
    for (int off = 16; off > 0; off >>= 1) {
        acc0 += __shfl_xor(acc0, off, 32);
        acc1 += __shfl_xor(acc1, off, 32);
        acc2 += __shfl_xor(acc2, off, 32);
        acc3 += __shfl_xor(acc3, off, 32);
    }
    if (lane == 0) {
        float* pr = partial + (size_t)blockIdx.x * HID + row0;
        pr[0] = acc0; pr[1] = acc1; pr[2] = acc2; pr[3] = acc3;
    }
}

// ---------------------------------------------------------------------------
// Kernel 2: fused { deterministic chunk reduction + b1 + ReLU } -> LDS ->
// layer-2 GEMV (256x256) + b2 + ReLU.  1 block x 256 threads.
// ---------------------------------------------------------------------------
__global__ void k_mid(const float* __restrict__ partial,
                      const float* __restrict__ b1,
                      const float* __restrict__ W2,
                      const float* __restrict__ b2,
                      float*       __restrict__ h2)
{
    __shared__ float sh[HID];
    const int t = threadIdx.x;

    float s = 0.f;
    for (int c = 0; c < NCHUNK; ++c) s += partial[c * HID + t];
    s += b1[t];
    sh[t] = fmaxf(s, 0.f);
    __syncthreads();

    const float* wr = W2 + (size_t)t * HID;
    float s2 = 0.f;
    #pragma unroll 4
    for (int k = 0; k < HID; k += 4) {
        s2 += wr[k+0]*sh[k+0] + wr[k+1]*sh[k+1] + wr[k+2]*sh[k+2] + wr[k+3]*sh[k+3];
    }
    s2 += b2[t];
    h2[t] = fmaxf(s2, 0.f);
}

// ---------------------------------------------------------------------------
// Kernel 3: layer 3 via V_WMMA_F32_16X16X4_F32, branch-free operand setup.
//   A (16x4)  = h2 broadcast to every row M:  lane<16 -> {K0,K1}, lane>=16 ->
//               {K2,K3}; one uniform ds_load_b64 per step, no lane masking.
//   B (4x16)  = W3 tile, row-K striped across lanes: lane n=lane&15 supplies
//               column n; per-lane global_load_b64 of W3[tile*16+n][kk+koff].
//   D (16x16) = every row M holds the same 16 outputs
//               D[m][n] = sum_k h2[k] * W3[tile*16+n][k];
//               C/D layout VGPR0: lanes0-15 = (M=0, N=lane) -> c[0] is the
//               answer for column n = lane on lanes 0..15.
// 1088 tiles; grid = 136 blocks x 256 threads (8 waves, 1 tile/wave).
// ---------------------------------------------------------------------------
__global__ void k_layer3_wmma(const float* __restrict__ h2,
                              const float* __restrict__ W3,
                              const float* __restrict__ b3,
                              float*       __restrict__ logits)
{
    __shared__ float sh[HID];
    sh[threadIdx.x] = h2[threadIdx.x];
    __syncthreads();

    const int lane = threadIdx.x & 31;
    const int wave = threadIdx.x >> 5;
    const int tile = blockIdx.x * 8 + wave;          // 0..1087
    const int n    = lane & 15;                      // output column within tile
    const int koff = (lane >> 4) << 1;               // half-wave K offset: 0 or 2
    const float* wr = W3 + (size_t)(tile * 16 + n) * HID + koff;

    v8f c = {0.f, 0.f, 0.f, 0.f, 0.f, 0.f, 0.f, 0.f};

    #pragma unroll 4
    for (int kk = 0; kk < HID; kk += 4) {
        const v2f a = *reinterpret_cast<const v2f*>(sh + kk + koff);   // uniform LDS b64
        const v2f b = *reinterpret_cast<const v2f*>(wr + kk);          // global b64
        c = __builtin_amdgcn_wmma_f32_16x16x4_f32(
                /*neg_a=*/false, a, /*neg_b=*/false, b,
                /*c_mod=*/(short)0, c, /*reuse_a=*/false, /*reuse_b=*/false);
    }

    if (lane < 16) {
        const int o = tile * 16 + n;
        logits[o] = c[0] + b3[o];
    }
}

// ---------------------------------------------------------------------------
// Kernel 4: softmax over rows of [1024,17], topology mask, renormalize with
// the reference's 1e-8 clamp semantics.
// ---------------------------------------------------------------------------
__global__ void k_softmax_mask(const float* __restrict__ logits,
                               const int*   __restrict__ topo,
                               const float* __restrict__ rmask,
                               float*       __restrict__ out)
{
    const int d = blockIdx.x * blockDim.x + threadIdx.x;
    if (d >= DD) return;
    const float* lr = logits + d * (NB + 1);

    float m = lr[0];
    #pragma unroll
    for (int j = 1; j < NB + 1; ++j) m = fmaxf(m, lr[j]);

    float em[NB + 1];
    float se = 0.f, sm = 0.f;
    #pragma unroll
    for (int j = 0; j < NB + 1; ++j) {
        const float msk = (j == 0) ? 1.0f : rmask[topo[d * NB + (j - 1)]];
        const float ev  = __expf(lr[j] - m);
        se += ev;
        const float evm = ev * msk;
        sm += evm;
        em[j] = evm;
    }
    // probs = e/se; masked = probs*mask; rs = max(sum(masked), 1e-8); out = masked/rs
    const float rs  = fmaxf(sm / se, 1e-8f);
    const float inv = 1.0f / (se * rs);
    #pragma unroll
    for (int j = 0; j < NB + 1; ++j) out[d * (NB + 1) + j] = em[j] * inv;
}

// ---------------------------------------------------------------------------
// Launch
// ---------------------------------------------------------------------------
extern "C" void kernel_launch(void* const* d_in, const int* in_sizes, int n_in,
                              void* d_out, int out_size, void* d_ws, size_t ws_size,
                              hipStream_t stream)
{
    const float* rep   = (const float*)d_in[0];
    const float* task  = (const float*)d_in[1];
    const int*   topo  = (const int*)  d_in[2];
    const float* rmask = (const float*)d_in[3];
    const float* W1    = (const float*)d_in[4];
    const float* b1    = (const float*)d_in[5];
    const float* W2    = (const float*)d_in[6];
    const float* b2    = (const float*)d_in[7];
    const float* W3    = (const float*)d_in[8];
    const float* b3    = (const float*)d_in[9];
    float* out = (float*)d_out;

    float* ws      = (float*)d_ws;
    float* partial = ws;                         // NCHUNK * 256
    float* h2      = partial + NCHUNK * HID;     // 256
    float* logits  = h2 + HID;                   // 17408
    (void)in_sizes; (void)n_in; (void)out_size; (void)ws_size;

    k_layer1      <<<dim3(NCHUNK, 8), 256, 0, stream>>>(rep, task, topo, rmask, W1, partial);
    k_mid         <<<1,   256, 0, stream>>>(partial, b1, W2, b2, h2);
    k_layer3_wmma <<<136, 256, 0, stream>>>(h2, W3, b3, logits);
    k_softmax_mask<<<4,   256, 0, stream>>>(logits, topo, rmask, out);
}